// KMeansLoss_17703855194727
// MI455X (gfx1250) — compile-verified
//
#include <hip/hip_runtime.h>
#include <hip/hip_bf16.h>
#include <math.h>

// ---------------------------------------------------------------------------
// CDNA5 (gfx1250) implementation of the kmeans-cluster + patch-discriminator
// + BCE pipeline. fp32 throughout (reference is fp32). conv2 (64->128, 3x3,
// stride2) is the compute-dense stage and is mapped to V_WMMA_F32_16X16X4_F32
// as 9 tap-wise rank-64 updates with linear addressing.
// ---------------------------------------------------------------------------

typedef __attribute__((ext_vector_type(2))) float v2f;
typedef __attribute__((ext_vector_type(4))) float v4f;
typedef __attribute__((ext_vector_type(8))) float v8f;

// ---------------------------------------------------------------------------
// K-means (2 clusters, 10 iterations, init = rows 0,1). One block per batch
// sample. Centers live in LDS. X is [C, D] row-major, D = H*W (D % 4 == 0).
// All bulk traffic uses b128 (float4) accesses.
// ---------------------------------------------------------------------------
template<int C, int D>
__global__ __launch_bounds__(256) void kmeans_kernel(const float* __restrict__ Xall,
                                                     float* __restrict__ imgAll)
{
    constexpr int D4 = D / 4;
    extern __shared__ float smem[];
    float* c0  = smem;              // [D]
    float* c1  = smem + D;          // [D]
    float* xsq = smem + 2 * D;      // [C]
    float* red = xsq + C;           // [16]
    int*   ids = (int*)(red + 16);  // [C]
    int*   cnt = ids + C;           // [2]

    const int b = blockIdx.x;
    const float* X = Xall + (size_t)b * C * D;
    const v4f*  X4 = (const v4f*)X;
    float* out = imgAll + (size_t)b * D;

    const int lane   = threadIdx.x & 31;
    const int wave   = threadIdx.x >> 5;
    const int nwaves = blockDim.x >> 5;

    // x_sq per row (wave-per-row, b128 loads)
    for (int i = wave; i < C; i += nwaves) {
        const v4f* row = X4 + (size_t)i * D4;
        float s = 0.f;
        for (int j = lane; j < D4; j += 32) {
            v4f v = row[j];
            s = fmaf(v[0], v[0], s); s = fmaf(v[1], v[1], s);
            s = fmaf(v[2], v[2], s); s = fmaf(v[3], v[3], s);
        }
        for (int off = 16; off; off >>= 1) s += __shfl_down(s, off, 32);
        if (lane == 0) xsq[i] = s;
    }
    // init centers = rows 0,1
    for (int j = threadIdx.x; j < D4; j += blockDim.x) {
        ((v4f*)c0)[j] = X4[j];
        ((v4f*)c1)[j] = X4[D4 + j];
    }
    __syncthreads();

    // 10 update iterations + 1 final assignment pass
    for (int it = 0; it <= 10; ++it) {
        // |c|^2 block reduction
        float p0 = 0.f, p1 = 0.f;
        for (int j = threadIdx.x; j < D4; j += blockDim.x) {
            v4f a = ((const v4f*)c0)[j];
            v4f c = ((const v4f*)c1)[j];
            p0 = fmaf(a[0], a[0], p0); p0 = fmaf(a[1], a[1], p0);
            p0 = fmaf(a[2], a[2], p0); p0 = fmaf(a[3], a[3], p0);
            p1 = fmaf(c[0], c[0], p1); p1 = fmaf(c[1], c[1], p1);
            p1 = fmaf(c[2], c[2], p1); p1 = fmaf(c[3], c[3], p1);
        }
        for (int off = 16; off; off >>= 1) {
            p0 += __shfl_down(p0, off, 32);
            p1 += __shfl_down(p1, off, 32);
        }
        if (lane == 0) { red[wave] = p0; red[8 + wave] = p1; }
        if (threadIdx.x == 0) { cnt[0] = 0; cnt[1] = 0; }
        __syncthreads();
        float csq0 = 0.f, csq1 = 0.f;
        for (int wv = 0; wv < nwaves; ++wv) { csq0 += red[wv]; csq1 += red[8 + wv]; }

        // assignment (wave-per-row dot products against LDS centers)
        for (int i = wave; i < C; i += nwaves) {
            const v4f* row = X4 + (size_t)i * D4;
            float d0 = 0.f, d1 = 0.f;
            for (int j = lane; j < D4; j += 32) {
                v4f v  = row[j];
                v4f a  = ((const v4f*)c0)[j];
                v4f c  = ((const v4f*)c1)[j];
                d0 = fmaf(v[0], a[0], d0); d0 = fmaf(v[1], a[1], d0);
                d0 = fmaf(v[2], a[2], d0); d0 = fmaf(v[3], a[3], d0);
                d1 = fmaf(v[0], c[0], d1); d1 = fmaf(v[1], c[1], d1);
                d1 = fmaf(v[2], c[2], d1); d1 = fmaf(v[3], c[3], d1);
            }
            for (int off = 16; off; off >>= 1) {
                d0 += __shfl_down(d0, off, 32);
                d1 += __shfl_down(d1, off, 32);
            }
            if (lane == 0) {
                float dist0 = xsq[i] - 2.f * d0 + csq0;
                float dist1 = xsq[i] - 2.f * d1 + csq1;
                int id = (dist1 < dist0) ? 1 : 0;   // argmin, ties -> 0
                ids[i] = id;
                atomicAdd(&cnt[id], 1);
            }
        }
        __syncthreads();

        if (it < 10) {
            const int n0 = cnt[0], n1 = cnt[1];
            const float inv0 = n0 > 0 ? 1.f / (float)n0 : 0.f;
            const float inv1 = n1 > 0 ? 1.f / (float)n1 : 0.f;
            // scatter-mean: thread-per-float4-column, loop over rows (coalesced b128)
            for (int j = threadIdx.x; j < D4; j += blockDim.x) {
                v4f s0 = {0.f, 0.f, 0.f, 0.f};
                v4f s1 = {0.f, 0.f, 0.f, 0.f};
                const v4f* col = X4 + j;
                for (int i = 0; i < C; ++i) {
                    v4f v = col[(size_t)i * D4];
                    float m = (ids[i] == 0) ? 1.f : 0.f;
                    v4f mv = {m, m, m, m};
                    s0 += v * mv;
                    s1 += v - v * mv;
                }
                v4f oc0 = ((const v4f*)c0)[j];
                v4f oc1 = ((const v4f*)c1)[j];
                v4f i0 = {inv0, inv0, inv0, inv0};
                v4f i1 = {inv1, inv1, inv1, inv1};
                ((v4f*)c0)[j] = n0 > 0 ? s0 * i0 : oc0;
                ((v4f*)c1)[j] = n1 > 0 ? s1 * i1 : oc1;
            }
            __syncthreads();
        }
    }

    const int obj = (cnt[1] > cnt[0]) ? 1 : 0;      // argmax, ties -> 0
    const v4f* cc = obj ? (const v4f*)c1 : (const v4f*)c0;
    for (int j = threadIdx.x; j < D4; j += blockDim.x) ((v4f*)out)[j] = cc[j];
}

// ---------------------------------------------------------------------------
// conv1: 1 -> 64, 3x3, stride 2, SAME (pad_before = 0, pad_after = 1),
// fused leaky-ReLU(0.2). Direct conv, one thread per output element.
// ---------------------------------------------------------------------------
template<int H, int W, int H1, int W1>
__global__ __launch_bounds__(256) void conv1_kernel(const float* __restrict__ img,
                                                    const float* __restrict__ w1,
                                                    const float* __restrict__ b1,
                                                    float* __restrict__ out1)
{
    const int total = 16 * 64 * H1 * W1;
    int idx = blockIdx.x * blockDim.x + threadIdx.x;
    if (idx >= total) return;
    int ox = idx % W1; int t = idx / W1;
    int oy = t % H1;   t /= H1;
    int oc = t % 64;   int b = t / 64;

    const float* im = img + (size_t)b * H * W;
    const float* wk = w1 + oc * 9;
    float acc = b1[oc];
    #pragma unroll
    for (int kh = 0; kh < 3; ++kh) {
        int iy = oy * 2 + kh;
        if (iy >= H) continue;
        #pragma unroll
        for (int kw = 0; kw < 3; ++kw) {
            int ix = ox * 2 + kw;
            if (ix >= W) continue;
            acc = fmaf(im[iy * W + ix], wk[kh * 3 + kw], acc);
        }
    }
    out1[idx] = acc > 0.f ? acc : 0.2f * acc;
}

// ---------------------------------------------------------------------------
// conv2: 64 -> 128, 3x3, stride 2, SAME, fused leaky-ReLU(0.2).
// Implicit GEMM as 9 tap-wise rank-64 updates on V_WMMA_F32_16X16X4_F32.
// One block = (batch, 16-oc tile); one wave owns a 16x16 (spatial x oc) tile.
// Per tap: (iy,ix) and validity computed once per lane; inner ic-loop has
// linear addressing (immediate offsets) and aligned ds_load_b64 for B.
// Fragment layouts (ISA 7.12.2):
//   A 16x4 f32 : lane&15 = M row; VGPR j holds K = (lane>=16 ? 2:0) + j
//   B 4x16 f32 : VGPR j holds K rows {j, j+2} striped over lane&15 = N col
//   C/D        : VGPR v -> M = v + (lane>=16 ? 8:0), N = lane&15
// B stored LDS-transposed Bt[n][tap*64+ic] with stride 580 dwords:
//   bank = (4n + ic + koff) mod 64  -> conflict-free b64 across 32 lanes.
// ---------------------------------------------------------------------------
template<int H1, int W1, int H2, int W2>
__global__ __launch_bounds__(256) void conv2_wmma_kernel(const float* __restrict__ in1,
                                                         const float* __restrict__ w2,
                                                         const float* __restrict__ b2,
                                                         float* __restrict__ out2)
{
    constexpr int M = H2 * W2;
    constexpr int K = 64 * 9;                // 576
    constexpr int numTiles = (M + 15) / 16;
    constexpr int BSTRIDE = 580;             // padded row stride (dwords)
    __shared__ float Bt[16 * BSTRIDE];       // 37.1 KB

    const int b      = blockIdx.x >> 3;
    const int ocBase = (blockIdx.x & 7) * 16;

    // stage B: coalesced read of w2, scattered LDS write (one time)
    // Bt[n*BSTRIDE + r*64 + ic] = w2[(ocBase+n)*576 + ic*9 + r]
    for (int t = threadIdx.x; t < 16 * K; t += blockDim.x) {
        int n  = t / K;
        int q  = t - n * K;          // q = ic*9 + r (linear in w2)
        int ic = q / 9;
        int r  = q - ic * 9;
        Bt[n * BSTRIDE + r * 64 + ic] = w2[(size_t)(ocBase + n) * K + q];
    }
    __syncthreads();

    const int lane     = threadIdx.x & 31;
    const int wave     = threadIdx.x >> 5;
    const int nCol     = lane & 15;           // N col for B/store, M row for A
    const int koffBase = (lane >> 4) * 2;     // upper half-wave holds K+2/K+3

    const float* inB = in1 + (size_t)b * 64 * H1 * W1;
    const float* bRow = &Bt[nCol * BSTRIDE + koffBase];

    for (int mt = wave; mt < numTiles; mt += 8) {
        int mrow = mt * 16 + nCol;
        int mc = mrow < M ? mrow : M - 1;     // clamp OOB rows (masked at store)
        int oy = mc / W2, ox = mc % W2;

        v8f acc = {0.f, 0.f, 0.f, 0.f, 0.f, 0.f, 0.f, 0.f};
        #pragma unroll
        for (int r = 0; r < 9; ++r) {
            const int kh = r / 3, kw = r - kh * 3;   // constants after unroll
            int iy = oy * 2 + kh;                     // pad_before = 0
            int ix = ox * 2 + kw;
            bool valid = (iy < H1) & (ix < W1);
            const float* aPtr = inB + ((valid ? (iy * W1 + ix) : 0) + koffBase * (H1 * W1));
            const float* bPtr = bRow + r * 64;

            #pragma unroll 4
            for (int ic = 0; ic < 64; ic += 4) {
                v2f a, bb;
                float a0 = aPtr[ic * (H1 * W1)];
                float a1 = aPtr[(ic + 1) * (H1 * W1)];
                a[0] = valid ? a0 : 0.f;
                a[1] = valid ? a1 : 0.f;
                bb = *(const v2f*)&bPtr[ic];           // aligned ds_load_b64
                acc = __builtin_amdgcn_wmma_f32_16x16x4_f32(
                    false, a, false, bb, (short)0, acc, false, false);
            }
        }

        const int oc = ocBase + nCol;
        const float bias = b2[oc];
        const int mOff = (lane >> 4) * 8;
        #pragma unroll
        for (int v = 0; v < 8; ++v) {
            int m = mt * 16 + mOff + v;
            if (m < M) {
                float res = acc[v] + bias;
                res = res > 0.f ? res : 0.2f * res;
                out2[((size_t)b * 128 + oc) * M + m] = res;
            }
        }
    }
}

// ---------------------------------------------------------------------------
// conv3: 128 -> 1, 3x3, stride 1, SAME (pad 1), + bias, then mean over
// spatial -> per-sample logit. One block per batch sample.
// ---------------------------------------------------------------------------
template<int H2, int W2>
__global__ __launch_bounds__(256) void conv3_mean_kernel(const float* __restrict__ out2,
                                                         const float* __restrict__ w3,
                                                         const float* __restrict__ b3,
                                                         float* __restrict__ logits)
{
    constexpr int M = H2 * W2;
    const int b = blockIdx.x;
    const float* in = out2 + (size_t)b * 128 * M;

    float local = 0.f;
    for (int p = threadIdx.x; p < M; p += blockDim.x) {
        int oy = p / W2, ox = p % W2;
        float acc = b3[0];
        for (int ic = 0; ic < 128; ++ic) {
            const float* ch = in + (size_t)ic * M;
            const float* wk = w3 + ic * 9;
            #pragma unroll
            for (int kh = 0; kh < 3; ++kh) {
                int iy = oy + kh - 1;
                if ((unsigned)iy >= (unsigned)H2) continue;
                #pragma unroll
                for (int kw = 0; kw < 3; ++kw) {
                    int ix = ox + kw - 1;
                    if ((unsigned)ix >= (unsigned)W2) continue;
                    acc = fmaf(ch[iy * W2 + ix], wk[kh * 3 + kw], acc);
                }
            }
        }
        local += acc;
    }
    __shared__ float red[8];
    const int lane = threadIdx.x & 31, wave = threadIdx.x >> 5;
    for (int off = 16; off; off >>= 1) local += __shfl_down(local, off, 32);
    if (lane == 0) red[wave] = local;
    __syncthreads();
    if (threadIdx.x == 0) {
        float s = 0.f;
        for (int i = 0; i < (int)(blockDim.x >> 5); ++i) s += red[i];
        logits[b] = s / (float)M;
    }
}

// ---------------------------------------------------------------------------
// BCE-with-logits, mean over 16 samples -> d_out[slot]
// ---------------------------------------------------------------------------
__global__ void bce_kernel(const float* __restrict__ logits,
                           const float* __restrict__ labels,
                           float* __restrict__ out, int slot)
{
    int i = threadIdx.x;
    float t = 0.f;
    if (i < 16) {
        float l = logits[i], y = labels[i];
        t = fmaxf(l, 0.f) - l * y + log1pf(expf(-fabsf(l)));
    }
    for (int off = 16; off; off >>= 1) t += __shfl_down(t, off, 32);
    if (i == 0) out[slot] = t / 16.f;
}

// ---------------------------------------------------------------------------
// Host-side per-map driver
// ---------------------------------------------------------------------------
template<int C, int H, int W>
static void run_map(const float* X,
                    const float* w1, const float* b1,
                    const float* w2, const float* b2,
                    const float* w3, const float* b3,
                    const float* labels, float* d_out, int slot,
                    float* img, float* o1, float* o2, float* logits,
                    hipStream_t stream)
{
    constexpr int D  = H * W;
    constexpr int H1 = H / 2, W1 = W / 2;
    constexpr int H2 = H1 / 2, W2 = W1 / 2;

    size_t smem = (size_t)(2 * D + C + 16 + C + 2) * sizeof(float);
    kmeans_kernel<C, D><<<16, 256, smem, stream>>>(X, img);

    const int total1 = 16 * 64 * H1 * W1;
    conv1_kernel<H, W, H1, W1><<<(total1 + 255) / 256, 256, 0, stream>>>(img, w1, b1, o1);

    conv2_wmma_kernel<H1, W1, H2, W2><<<16 * 8, 256, 0, stream>>>(o1, w2, b2, o2);

    conv3_mean_kernel<H2, W2><<<16, 256, 0, stream>>>(o2, w3, b3, logits);

    bce_kernel<<<1, 32, 0, stream>>>(logits, labels, d_out, slot);
}

extern "C" void kernel_launch(void* const* d_in, const int* in_sizes, int n_in,
                              void* d_out, int out_size, void* d_ws, size_t ws_size,
                              hipStream_t stream)
{
    const float* lf     = (const float*)d_in[0];
    const float* mf     = (const float*)d_in[1];
    const float* sf     = (const float*)d_in[2];
    const float* labels = (const float*)d_in[5];

    const float* lw1 = (const float*)d_in[7];  const float* lb1 = (const float*)d_in[8];
    const float* lw2 = (const float*)d_in[9];  const float* lb2 = (const float*)d_in[10];
    const float* lw3 = (const float*)d_in[11]; const float* lb3 = (const float*)d_in[12];

    const float* mw1 = (const float*)d_in[13]; const float* mb1 = (const float*)d_in[14];
    const float* mw2 = (const float*)d_in[15]; const float* mb2 = (const float*)d_in[16];
    const float* mw3 = (const float*)d_in[17]; const float* mb3 = (const float*)d_in[18];

    const float* sw1 = (const float*)d_in[19]; const float* sb1 = (const float*)d_in[20];
    const float* sw2 = (const float*)d_in[21]; const float* sb2 = (const float*)d_in[22];
    const float* sw3 = (const float*)d_in[23]; const float* sb3 = (const float*)d_in[24];

    float* out = (float*)d_out;

    // Workspace layout (reused sequentially by all three maps; max = lf sizes):
    //   img    @ 0         :  16*6400      floats =   409,600 B
    //   out1   @ 409600    :  16*64*1600   floats = 6,553,600 B
    //   out2   @ 6963200   :  16*128*400   floats = 3,276,800 B
    //   logits @ 10240000  :  16           floats
    char* ws = (char*)d_ws;
    float* img    = (float*)(ws + 0);
    float* o1     = (float*)(ws + 409600);
    float* o2     = (float*)(ws + 6963200);
    float* logits = (float*)(ws + 10240000);

    run_map<256, 80, 80>(lf, lw1, lb1, lw2, lb2, lw3, lb3, labels, out, 0,
                         img, o1, o2, logits, stream);
    run_map<512, 40, 40>(mf, mw1, mb1, mw2, mb2, mw3, mb3, labels, out, 1,
                         img, o1, o2, logits, stream);
    run_map<1024, 20, 20>(sf, sw1, sb1, sw2, sb2, sw3, sb3, labels, out, 2,
                          img, o1, o2, logits, stream);
}